// SpGraphAttentionLayer_28475633173127
// MI455X (gfx1250) — compile-verified
//
#include <hip/hip_runtime.h>
#include <hip/hip_bf16.h>

typedef __attribute__((ext_vector_type(2))) float v2f;
typedef __attribute__((ext_vector_type(8))) float v8f;

#define GAT_B 4
#define GAT_N 8192
#define GAT_E 131072
#define GAT_F 256
#define GAT_BN (GAT_B * GAT_N)       // 32768 flattened rows
#define ALPHA_SLOPE 0.2f
#define EPS_VAL 0.001f

// ---------------------------------------------------------------------------
// Kernel 0: zero the accumulators (hp0 and rowsum) — we atomically accumulate
// into them each launch, and the harness does not re-zero workspace.
// ---------------------------------------------------------------------------
__global__ void gat_zero(float* __restrict__ hp, float* __restrict__ rowsum) {
  size_t i = (size_t)blockIdx.x * blockDim.x + threadIdx.x;
  if (i < (size_t)GAT_BN * GAT_F) hp[i] = 0.0f;
  if (i < (size_t)GAT_BN) rowsum[i] = 0.0f;
}

// ---------------------------------------------------------------------------
// Kernel 1: h = x @ W via V_WMMA_F32_16X16X4_F32 (fp32-exact WMMA path).
// Grid: 2048 blocks (one 16-row tile of the flattened 32768x256 problem each),
// 128 threads = 4 waves; wave w owns output columns [64w, 64w+64).
// Per K-step of 4: one float2 A load, eight B loads, four WMMAs.
// ---------------------------------------------------------------------------
__global__ void __launch_bounds__(128)
gat_gemm_wmma(const float* __restrict__ x, const float* __restrict__ W,
              float* __restrict__ h) {
  const int lane   = threadIdx.x & 31;
  const int wave   = threadIdx.x >> 5;     // 0..3 -> 64-col group
  const int half   = lane >> 4;            // 0: K pair {0,1}, 1: K pair {2,3}
  const int lm     = lane & 15;            // M (for A) / N (for B,C,D) sublane
  const int rowT   = blockIdx.x * 16;      // row tile base in flattened B*N
  const int n0     = wave * 64;

  // A: lane holds x[rowT+lm, k + 2*half .. +1]  (contiguous float2)
  const float* arow = x + (size_t)(rowT + lm) * GAT_F + half * 2;
  // B: lane holds W[k + 2*half][n0 + lm + {0,16,32,48}] and next-K row
  const float* bcol = W + n0 + lm;

  v8f c0 = {}, c1 = {}, c2 = {}, c3 = {};
  for (int k = 0; k < GAT_F; k += 4) {
    v2f a;
    a.x = arow[k];
    a.y = arow[k + 1];
    const float* bk = bcol + (size_t)(k + half * 2) * GAT_F;
    v2f b0, b1, b2, b3;
    b0.x = bk[0];           b0.y = bk[GAT_F];
    b1.x = bk[16];          b1.y = bk[GAT_F + 16];
    b2.x = bk[32];          b2.y = bk[GAT_F + 32];
    b3.x = bk[48];          b3.y = bk[GAT_F + 48];
    c0 = __builtin_amdgcn_wmma_f32_16x16x4_f32(false, a, false, b0, (short)0, c0, false, false);
    c1 = __builtin_amdgcn_wmma_f32_16x16x4_f32(false, a, false, b1, (short)0, c1, false, false);
    c2 = __builtin_amdgcn_wmma_f32_16x16x4_f32(false, a, false, b2, (short)0, c2, false, false);
    c3 = __builtin_amdgcn_wmma_f32_16x16x4_f32(false, a, false, b3, (short)0, c3, false, false);
  }

  // D layout: VGPR r holds M = r (lanes 0-15) / M = r+8 (lanes 16-31), N = lm
  float* hrow = h + (size_t)rowT * GAT_F + n0;
#pragma unroll
  for (int r = 0; r < 8; ++r) {
    const int m = r + half * 8;
    hrow[(size_t)m * GAT_F + lm +  0] = c0[r];
    hrow[(size_t)m * GAT_F + lm + 16] = c1[r];
    hrow[(size_t)m * GAT_F + lm + 32] = c2[r];
    hrow[(size_t)m * GAT_F + lm + 48] = c3[r];
  }
}

// ---------------------------------------------------------------------------
// Kernel 2: s = h @ a1, t = h @ a2. One wave per node, coalesced strided
// loads, wave32 butterfly reduction.
// ---------------------------------------------------------------------------
__global__ void __launch_bounds__(256)
gat_scores(const float* __restrict__ h, const float* __restrict__ avec,
           float* __restrict__ s, float* __restrict__ t) {
  const int node = blockIdx.x * 8 + (threadIdx.x >> 5);
  const int lane = threadIdx.x & 31;
  const float* hr = h + (size_t)node * GAT_F;
  float sv = 0.0f, tv = 0.0f;
#pragma unroll
  for (int j = 0; j < GAT_F; j += 32) {
    const float hv = hr[j + lane];
    sv += hv * avec[j + lane];
    tv += hv * avec[GAT_F + j + lane];
  }
#pragma unroll
  for (int off = 16; off > 0; off >>= 1) {
    sv += __shfl_xor(sv, off, 32);
    tv += __shfl_xor(tv, off, 32);
  }
  if (lane == 0) { s[node] = sv; t[node] = tv; }
}

// ---------------------------------------------------------------------------
// Kernel 3: per-edge attention numerator + rowsum accumulation.
// e = exp(-leaky_relu(s[src] + t[dst])); rowsum[src] += e (global f32 atomic).
// ---------------------------------------------------------------------------
__global__ void __launch_bounds__(256)
gat_edge(const long long* __restrict__ adj, const float* __restrict__ s,
         const float* __restrict__ t, float* __restrict__ e,
         float* __restrict__ rowsum) {
  const int i = blockIdx.x * blockDim.x + threadIdx.x;   // 0 .. B*E-1
  if (i >= GAT_B * GAT_E) return;
  const int b = i / GAT_E;
  const int k = i - b * GAT_E;
  const long long src = adj[(size_t)b * GAT_E + k];                  // adj[0][b][k]
  const long long dst = adj[(size_t)(GAT_B + b) * GAT_E + k];        // adj[1][b][k]
  const float sc = s[(size_t)b * GAT_N + src] + t[(size_t)b * GAT_N + dst];
  const float lr = sc > 0.0f ? sc : ALPHA_SLOPE * sc;
  const float ev = __expf(-lr);
  e[i] = ev;
  atomicAdd(&rowsum[(size_t)b * GAT_N + src], ev);
}

// ---------------------------------------------------------------------------
// Kernel 4: scatter hp0[src,:] += e * h[dst,:]. 64 lanes per edge, float4 per
// lane -> each edge issues a fully coalesced 1KB read + 1KB of f32 atomics
// that resolve in L2's atomic units (whole hp0 is L2-resident).
// ---------------------------------------------------------------------------
__global__ void __launch_bounds__(256)
gat_scatter(const long long* __restrict__ adj, const float* __restrict__ e,
            const float* __restrict__ h, float* __restrict__ hp) {
  const int eidx = blockIdx.x * 4 + (threadIdx.x >> 6);  // 4 edges per block
  const int j    = (threadIdx.x & 63) * 4;               // feature offset
  const int b = eidx / GAT_E;
  const int k = eidx - b * GAT_E;
  const long long src = adj[(size_t)b * GAT_E + k];
  const long long dst = adj[(size_t)(GAT_B + b) * GAT_E + k];
  const float ev = e[eidx];
  const float4 hv = *(const float4*)(h + ((size_t)b * GAT_N + dst) * GAT_F + j);
  float* out = hp + ((size_t)b * GAT_N + src) * GAT_F + j;
  atomicAdd(out + 0, ev * hv.x);
  atomicAdd(out + 1, ev * hv.y);
  atomicAdd(out + 2, ev * hv.z);
  atomicAdd(out + 3, ev * hv.w);
}

// ---------------------------------------------------------------------------
// Kernel 5: out = elu(hp0 / (rowsum + eps))
// ---------------------------------------------------------------------------
__global__ void __launch_bounds__(256)
gat_final(const float* __restrict__ hp, const float* __restrict__ rowsum,
          float* __restrict__ out) {
  const size_t i = (size_t)blockIdx.x * blockDim.x + threadIdx.x;
  const size_t node = i >> 8;  // /GAT_F
  const float v = hp[i] / (rowsum[node] + EPS_VAL);
  out[i] = v > 0.0f ? v : (__expf(v) - 1.0f);
}

// ---------------------------------------------------------------------------
extern "C" void kernel_launch(void* const* d_in, const int* in_sizes, int n_in,
                              void* d_out, int out_size, void* d_ws, size_t ws_size,
                              hipStream_t stream) {
  const float*     x    = (const float*)d_in[0];       // (B, N, F_IN) f32
  const long long* adj  = (const long long*)d_in[1];   // (2, B, E)   i64
  const float*     W    = (const float*)d_in[2];       // (F_IN, F_OUT) f32
  const float*     avec = (const float*)d_in[3];       // (1, 2*F_OUT)  f32
  float*           out  = (float*)d_out;               // (B, N, F_OUT) f32

  // Workspace carve-up
  char* ws = (char*)d_ws;
  const size_t hBytes  = (size_t)GAT_BN * GAT_F * sizeof(float);   // 32 MB
  float* h      = (float*)(ws);                 ws += hBytes;
  float* hp     = (float*)(ws);                 ws += hBytes;      // 32 MB
  float* s      = (float*)(ws);                 ws += (size_t)GAT_BN * sizeof(float);
  float* t      = (float*)(ws);                 ws += (size_t)GAT_BN * sizeof(float);
  float* rowsum = (float*)(ws);                 ws += (size_t)GAT_BN * sizeof(float);
  float* evec   = (float*)(ws);                 ws += (size_t)GAT_B * GAT_E * sizeof(float);

  // 0) zero accumulators
  gat_zero<<<(GAT_BN * GAT_F + 255) / 256, 256, 0, stream>>>(hp, rowsum);
  // 1) h = x @ W  (fp32 WMMA)
  gat_gemm_wmma<<<GAT_BN / 16, 128, 0, stream>>>(x, W, h);
  // 2) s, t projections
  gat_scores<<<GAT_BN / 8, 256, 0, stream>>>(h, avec, s, t);
  // 3) edge weights + rowsum
  gat_edge<<<(GAT_B * GAT_E + 255) / 256, 256, 0, stream>>>(adj, s, t, evec, rowsum);
  // 4) weighted scatter
  gat_scatter<<<(GAT_B * GAT_E) / 4, 256, 0, stream>>>(adj, evec, h, hp);
  // 5) normalize + ELU
  gat_final<<<(GAT_BN * GAT_F) / 256, 256, 0, stream>>>(hp, rowsum, out);
}